// GTN_78546361909729
// MI455X (gfx1250) — compile-verified
//
#include <hip/hip_runtime.h>

// GTN forward, B=32, N=256, E=5, C=4, L from w_rest size.
// Heavy math: 2 x 128 batched 256^3 f32 GEMMs via V_WMMA_F32_16X16X4_F32,
// with double-buffered GLOBAL_LOAD_ASYNC_TO_LDS_B128 staging (ASYNCcnt).
// Workspace (floats): a/q | b/normH | H | deg_inv | softmax weights  (~96.2 MiB)

typedef __attribute__((ext_vector_type(2))) float v2f;
typedef __attribute__((ext_vector_type(4))) float v4f;
typedef __attribute__((ext_vector_type(8))) float v8f;

static constexpr int BB = 32;    // batch
static constexpr int Nn = 256;   // nodes
static constexpr int EE = 5;     // edge types
static constexpr int CC = 4;     // channels
static constexpr int TM = 128;   // 128x128 block tile
static constexpr int KC = 32;    // K chunk
static constexpr int AS = 36;    // A-tile LDS stride (32+4): conflict-free b64 frag reads
static constexpr int BS = 136;   // B-tile LDS stride (128+8): splits lane halves across banks

// ---------------- K0: softmax of all weight rows -----------------------------
__global__ void gtn_softmax(const float* __restrict__ w1,
                            const float* __restrict__ w2,
                            const float* __restrict__ wr,
                            float* __restrict__ sw, int layers) {
  int rows = 8 + 4 * layers;
  for (int t = threadIdx.x; t < rows; t += blockDim.x) {
    const float* src;
    if (t < 4)      src = w1 + t * EE;
    else if (t < 8) src = w2 + (t - 4) * EE;
    else            src = wr + (t - 8) * EE;
    float m = src[0];
#pragma unroll
    for (int e = 1; e < EE; ++e) m = fmaxf(m, src[e]);
    float ex[EE]; float s = 0.f;
#pragma unroll
    for (int e = 0; e < EE; ++e) { ex[e] = __expf(src[e] - m); s += ex[e]; }
    float inv = 1.f / s;
#pragma unroll
    for (int e = 0; e < EE; ++e) sw[t * EE + e] = ex[e] * inv;
  }
}

// ---------------- K1: fused gtconv producing a and b -------------------------
__global__ __launch_bounds__(256)
void gtn_conv_first(const float* __restrict__ A, const float* __restrict__ sw,
                    float* __restrict__ oa, float* __restrict__ ob) {
  const size_t t = (size_t)blockIdx.x * blockDim.x + threadIdx.x;
  const size_t total4 = (size_t)BB * Nn * Nn / 4;
  if (t >= total4) return;
  const size_t base = t * 4;
  const size_t b    = base >> 16;
  const size_t nm   = base & 65535;
  const float* ap = A + base * EE;          // 80B/thread, 16B aligned
  float v[20];
#pragma unroll
  for (int i = 0; i < 5; ++i) {
    v4f x = *(const v4f*)(ap + i * 4);
    v[i*4+0] = x[0]; v[i*4+1] = x[1]; v[i*4+2] = x[2]; v[i*4+3] = x[3];
  }
#pragma unroll
  for (int c = 0; c < CC; ++c) {
    v4f ra = {}, rb = {};
#pragma unroll
    for (int s = 0; s < 4; ++s) {
      float sa = 0.f, sb = 0.f;
#pragma unroll
      for (int e = 0; e < EE; ++e) {
        float x = v[s * EE + e];
        sa += sw[c * EE + e] * x;
        sb += sw[20 + c * EE + e] * x;
      }
      ra[s] = sa; rb[s] = sb;
    }
    size_t off = (((b * CC) + c) << 16) + nm;
    *(v4f*)(oa + off) = ra;
    *(v4f*)(ob + off) = rb;
  }
}

// Single-output gtconv for later layers.
__global__ __launch_bounds__(256)
void gtn_conv_one(const float* __restrict__ A, const float* __restrict__ sw,
                  float* __restrict__ oq) {
  const size_t t = (size_t)blockIdx.x * blockDim.x + threadIdx.x;
  const size_t total4 = (size_t)BB * Nn * Nn / 4;
  if (t >= total4) return;
  const size_t base = t * 4;
  const size_t b    = base >> 16;
  const size_t nm   = base & 65535;
  const float* ap = A + base * EE;
  float v[20];
#pragma unroll
  for (int i = 0; i < 5; ++i) {
    v4f x = *(const v4f*)(ap + i * 4);
    v[i*4+0] = x[0]; v[i*4+1] = x[1]; v[i*4+2] = x[2]; v[i*4+3] = x[3];
  }
#pragma unroll
  for (int c = 0; c < CC; ++c) {
    v4f rq = {};
#pragma unroll
    for (int s = 0; s < 4; ++s) {
      float sq = 0.f;
#pragma unroll
      for (int e = 0; e < EE; ++e) sq += sw[c * EE + e] * v[s * EE + e];
      rq[s] = sq;
    }
    *(v4f*)(oq + (((b * CC) + c) << 16) + nm) = rq;
  }
}

// ---------------- K3: row degrees of norm_add(H) -----------------------------
__global__ __launch_bounds__(256)
void gtn_deg(const float* __restrict__ H, float* __restrict__ deg_inv) {
  int row_id = blockIdx.x * 8 + (threadIdx.x >> 5);
  int lane   = threadIdx.x & 31;
  int mat    = row_id >> 8;
  int n      = row_id & 255;
  const float* hr = H + (size_t)mat * Nn * Nn + (size_t)n * Nn;
  float s = 0.f;
#pragma unroll
  for (int m = lane; m < Nn; m += 32) {
    float v = hr[m];
    s += (m == n) ? 1.0f : v;
  }
#pragma unroll
  for (int off = 16; off > 0; off >>= 1) s += __shfl_down(s, off, 32);
  if (lane == 0) deg_inv[row_id] = (s == 0.f) ? 0.f : 1.f / s;
}

// ---------------- K3b: Hn = norm_add(H) (diag->1, row-scale) -----------------
__global__ __launch_bounds__(256)
void gtn_normh(const float* __restrict__ H, const float* __restrict__ deg_inv,
               float* __restrict__ Hn) {
  const size_t t = (size_t)blockIdx.x * blockDim.x + threadIdx.x;
  const size_t total4 = (size_t)BB * CC * Nn * Nn / 4;
  if (t >= total4) return;
  const size_t base = t * 4;
  const size_t mat  = base >> 16;
  const int row  = (int)((base >> 8) & 255);
  const int col0 = (int)(base & 255);
  const float dv = deg_inv[mat * Nn + row];
  v4f x = *(const v4f*)(H + base);
#pragma unroll
  for (int u = 0; u < 4; ++u) {
    float e = ((col0 + u) == row) ? 1.0f : x[u];
    x[u] = e * dv;
  }
  *(v4f*)(Hn + base) = x;
}

// ---------------- K2/K4: batched WMMA GEMM, async double-buffered ------------
// 128 threads (4 waves), 128x128 tile, 64x64 per wave, 4x4 v8f accumulators.
// Global->LDS staging via GLOBAL_LOAD_ASYNC_TO_LDS_B128 + s_wait_asynccnt.
__global__ __launch_bounds__(128)
void gtn_matmul(const float* __restrict__ Am, const float* __restrict__ Bm,
                float* __restrict__ Om) {
  __shared__ float Asm[2][TM * AS];   // 128 x 32 per buffer (M x K)
  __shared__ float Bsm[2][KC * BS];   // 32 x 128 per buffer (K x N)

  const int mat = blockIdx.z;
  const int ti0 = blockIdx.y * TM;
  const int tj0 = blockIdx.x * TM;

  const float* Ab = Am + (size_t)mat * Nn * Nn;
  const float* Bb = Bm + (size_t)mat * Nn * Nn;
  float*       Ob = Om + (size_t)mat * Nn * Nn;

  const int tid  = threadIdx.x;
  const int lane = tid & 31;
  const int wv   = tid >> 5;
  const int wi   = (wv >> 1) * 64;
  const int wj   = (wv & 1) * 64;
  const int ml   = lane & 15;
  const int kb   = (lane >> 4) * 2;

  // Issue this thread's 16 async b128 copies (8 A-tile + 8 B-tile) for a chunk.
  auto issue = [&](int buf, int k0) {
#pragma unroll
    for (int it = 0; it < 8; ++it) {
      int idx = it * 128 + tid;                 // float4 slot 0..1023
      int r   = idx >> 3;                       // A row 0..127
      int c4  = (idx & 7) * 4;                  // A col
      unsigned lds = (unsigned)(unsigned long long)(const void*)&Asm[buf][r * AS + c4];
      unsigned long long g =
          (unsigned long long)(const void*)(Ab + (size_t)(ti0 + r) * Nn + k0 + c4);
      asm volatile("global_load_async_to_lds_b128 %0, %1, off"
                   :: "v"(lds), "v"(g) : "memory");
    }
#pragma unroll
    for (int it = 0; it < 8; ++it) {
      int idx = it * 128 + tid;
      int r   = idx >> 5;                       // B row 0..31
      int c4  = (idx & 31) * 4;                 // B col
      unsigned lds = (unsigned)(unsigned long long)(const void*)&Bsm[buf][r * BS + c4];
      unsigned long long g =
          (unsigned long long)(const void*)(Bb + (size_t)(k0 + r) * Nn + tj0 + c4);
      asm volatile("global_load_async_to_lds_b128 %0, %1, off"
                   :: "v"(lds), "v"(g) : "memory");
    }
  };

  v8f acc[4][4];
  v8f vzero = {};
#pragma unroll
  for (int i = 0; i < 4; ++i)
#pragma unroll
    for (int j = 0; j < 4; ++j) acc[i][j] = vzero;

  issue(0, 0);                                  // prologue: chunk 0 -> buffer 0

  constexpr int NCH = Nn / KC;                  // 8 chunks
  for (int c = 0; c < NCH; ++c) {
    const int buf = c & 1;
    if (c + 1 < NCH) {
      issue(buf ^ 1, (c + 1) * KC);             // prefetch next chunk
      // 16 newest copies may remain outstanding; chunk c's 16 are complete.
      asm volatile("s_wait_asynccnt 0x10" ::: "memory");
    } else {
      asm volatile("s_wait_asynccnt 0x0" ::: "memory");
    }
    __syncthreads();                            // all waves' tiles visible

    const float* As = Asm[buf];
    const float* Bs = Bsm[buf];
#pragma unroll
    for (int s = 0; s < KC / 4; ++s) {
      v2f af[4]; v2f bf[4];
#pragma unroll
      for (int i = 0; i < 4; ++i)               // A frag: A[m][kb..kb+1] (b64)
        af[i] = *(const v2f*)(&As[(wi + i * 16 + ml) * AS + s * 4 + kb]);
#pragma unroll
      for (int j = 0; j < 4; ++j) {             // B frag: B[kb..kb+1][n]
        int col = wj + j * 16 + ml;
        v2f bv;
        bv[0] = Bs[(s * 4 + kb + 0) * BS + col];
        bv[1] = Bs[(s * 4 + kb + 1) * BS + col];
        bf[j] = bv;
      }
#pragma unroll
      for (int i = 0; i < 4; ++i)
#pragma unroll
        for (int j = 0; j < 4; ++j)
          acc[i][j] = __builtin_amdgcn_wmma_f32_16x16x4_f32(
              false, af[i], false, bf[j], (short)0, acc[i][j], false, false);
    }
    __syncthreads();                            // done reading buf: safe to refill
  }

  const int rofs = (lane >> 4) * 8;             // D: m = r + 8*(lane>>4), n = lane&15
  const int nc   = lane & 15;
#pragma unroll
  for (int i = 0; i < 4; ++i)
#pragma unroll
    for (int j = 0; j < 4; ++j)
#pragma unroll
      for (int r = 0; r < 8; ++r) {
        int row = ti0 + wi + i * 16 + rofs + r;
        int col = tj0 + wj + j * 16 + nc;
        Ob[(size_t)row * Nn + col] = acc[i][j][r];
      }
}

// ---------------------------------------------------------------------------
extern "C" void kernel_launch(void* const* d_in, const int* in_sizes, int n_in,
                              void* d_out, int out_size, void* d_ws, size_t ws_size,
                              hipStream_t stream) {
  const float* A  = (const float*)d_in[0];   // (B,N,N,E)
  const float* w1 = (const float*)d_in[1];   // (C,E)
  const float* w2 = (const float*)d_in[2];   // (C,E)
  const float* wr = (const float*)d_in[3];   // (L-1,C,E)
  float* out = (float*)d_out;                // (B,C,N,N)

  const int layers = in_sizes[3] / (CC * EE);

  const size_t MATS = (size_t)BB * CC;       // 128
  const size_t ME   = (size_t)Nn * Nn;       // 65536
  float* ws   = (float*)d_ws;
  float* rA   = ws;                 // conv a, then per-layer q
  float* rB   = rA + MATS * ME;     // conv b, then per-layer normalized H
  float* rH   = rB + MATS * ME;     // H
  float* rDeg = rH + MATS * ME;     // B*C*N degree inverses
  float* rSW  = rDeg + MATS * Nn;   // softmaxed weights

  const unsigned convGrid = (unsigned)((BB * Nn * Nn / 4 + 255) / 256);
  const unsigned ewGrid   = (unsigned)((MATS * ME / 4 + 255) / 256);
  dim3 grid(Nn / TM, Nn / TM, (unsigned)MATS);

  gtn_softmax<<<1, 64, 0, stream>>>(w1, w2, wr, rSW, layers);
  gtn_conv_first<<<convGrid, 256, 0, stream>>>(A, rSW, rA, rB);
  gtn_matmul<<<grid, 128, 0, stream>>>(rA, rB, rH);          // H = a @ b

  for (int li = 0; li < layers; ++li) {
    gtn_deg<<<(unsigned)(MATS * Nn / 8), 256, 0, stream>>>(rH, rDeg);
    gtn_normh<<<ewGrid, 256, 0, stream>>>(rH, rDeg, rB);     // rB = norm_add(H)
    gtn_conv_one<<<convGrid, 256, 0, stream>>>(A, rSW + (8 + 4 * li) * EE, rA);
    float* dst = (li == layers - 1) ? out : rH;              // rH dead after normh
    gtn_matmul<<<grid, 128, 0, stream>>>(rB, rA, dst);       // H' = Hn @ q
  }
}